// MimickedSelfContactLoss_45664092291589
// MI455X (gfx1250) — compile-verified
//
#include <hip/hip_runtime.h>
#include <hip/hip_bf16.h>

typedef __attribute__((ext_vector_type(2))) float v2f;
typedef __attribute__((ext_vector_type(8))) float v8f;

#define CMAX 1024   // C = 1024 contact vertices in this problem

// ---------------------------------------------------------------------------
// Kernel 1: gather contact vertices into a dense [C,4] f32 array (x,y,z,0),
// precompute squared norms, and zero the scalar output accumulator.
// ---------------------------------------------------------------------------
__global__ void gather_contacts_kernel(const int* __restrict__ pc,
                                       const float* __restrict__ verts,
                                       float4* __restrict__ Xc,
                                       float* __restrict__ sq,
                                       float* __restrict__ out,
                                       int C) {
    int t = blockIdx.x * blockDim.x + threadIdx.x;
    if (t == 0) out[0] = 0.0f;               // output poisoned by harness -> zero it
    if (t < C) {
        int idx = pc[t];
        float x = verts[idx * 3 + 0];
        float y = verts[idx * 3 + 1];
        float z = verts[idx * 3 + 2];
        Xc[t] = make_float4(x, y, z, 0.0f);  // K padded 3->4 with zero
        sq[t] = x * x + y * y + z * z;
    }
}

// ---------------------------------------------------------------------------
// Kernel 2: one 128-thread block (4 waves) per 16-row tile of the C x C
// contact matrix. Gathered coords/norms/ids staged in LDS (24 KB of the
// 320 KB WGP pool). Each wave computes Gram tiles for a strided subset of
// J via V_WMMA_F32_16X16X4_F32 and tracks argmin on masked squared
// distance (monotonic in score = dist+1); sqrt/tanh only at the end.
//
// WMMA f32 16x16x4 operand layout (wave32, ISA 7.12.2):
//   A (16x4):  lane holds row M = lane%16; VGPR v holds K = 2*(lane/16)+v
//   B (4x16):  lane holds col N = lane%16; VGPR v holds K = 2*(lane/16)+v
//   D (16x16): VGPR v holds M = v + 8*(lane/16), N = lane%16
// ---------------------------------------------------------------------------
__global__ __launch_bounds__(128) void contact_argmin_kernel(
        const int* __restrict__ pc,
        const unsigned char* __restrict__ geomask,   // [N,N] bool bytes
        const float4* __restrict__ Xc,               // [C,4] packed coords
        const float* __restrict__ sq,                // [C]   squared norms
        float* __restrict__ out,
        int C, int N, float inv_C) {
    const int tid  = threadIdx.x;
    const int wave = tid >> 5;           // 0..3
    const int lane = tid & 31;           // wave32 lane
    const int half = lane >> 4;          // 0 or 1
    const int n16  = lane & 15;
    const int I    = blockIdx.x;         // row tile index
    const int rowBase = I * 16;

    __shared__ float sXc[CMAX * 4];      // 16 KB: packed (x,y,z,0)
    __shared__ float sSq[CMAX];          //  4 KB
    __shared__ int   sPc[CMAX];          //  4 KB
    __shared__ float redS[4 * 16];       // per-wave per-row best d^2
    __shared__ int   redJ[4 * 16];       // per-wave per-row best j

    // Cooperative stage of all gathered contact data into LDS.
    for (int t = tid; t < C; t += 128) {
        float4 x = Xc[t];
        ((float4*)sXc)[t] = x;
        sSq[t] = sq[t];
        sPc[t] = pc[t];
    }
    __syncthreads();

    // A operand (rows of this I-tile), reused across every J tile.
    v2f a;
    {
        const float* p = sXc + (rowBase + n16) * 4 + 2 * half;
        a.x = p[0];
        a.y = p[1];
    }

    // Per-output-register row metadata: row m = v + 8*half.
    float sqI[8];
    const unsigned char* mrow[8];
#pragma unroll
    for (int v = 0; v < 8; ++v) {
        int r = rowBase + 8 * half + v;
        sqI[v]  = sSq[r];
        mrow[v] = geomask + (size_t)sPc[r] * (size_t)N;
    }

    const float INF = __builtin_inff();
    float bestS[8];                      // masked squared distance
    int   bestJ[8];
#pragma unroll
    for (int v = 0; v < 8; ++v) { bestS[v] = INF; bestJ[v] = 0x7fffffff; }

    const int JT = C >> 4;               // 64 column tiles
    for (int J = wave; J < JT; J += 4) { // waves split the J range
        v2f b;
        const float* p = sXc + (J * 16 + n16) * 4 + 2 * half;
        b.x = p[0];
        b.y = p[1];

        // 16x16 Gram tile: D = A * B (+0), K = 4 (last K is the zero pad).
        v8f acc = {};
        acc = __builtin_amdgcn_wmma_f32_16x16x4_f32(
                /*neg_a=*/false, a, /*neg_b=*/false, b,
                /*c_mod=*/(short)0, acc, /*reuse_a=*/false, /*reuse_b=*/false);

        const int   j   = J * 16 + n16;  // absolute column (contact index)
        const float sqJ = sSq[j];
        const int   pcJ = sPc[j];

#pragma unroll
        for (int v = 0; v < 8; ++v) {
            float d2 = fmaxf(sqI[v] + sqJ - 2.0f * acc[v], 0.0f);
            float s  = (mrow[v][pcJ] != 0) ? d2 : INF;
            // strict < : smallest j wins ties within this lane's ascending columns
            if (s < bestS[v]) { bestS[v] = s; bestJ[v] = j; }
        }
    }

    // Cross-lane argmin within each 16-lane half (xor masks 1..8 stay
    // in-half); explicit tie-break on smaller j matches jnp.argmin.
#pragma unroll
    for (int off = 1; off < 16; off <<= 1) {
#pragma unroll
        for (int v = 0; v < 8; ++v) {
            float s2 = __shfl_xor(bestS[v], off, 32);
            int   j2 = __shfl_xor(bestJ[v], off, 32);
            if (s2 < bestS[v] || (s2 == bestS[v] && j2 < bestJ[v])) {
                bestS[v] = s2; bestJ[v] = j2;
            }
        }
    }

    // Owner lanes (n16==0 in each half) publish this wave's 16-row results.
    if (n16 == 0) {
#pragma unroll
        for (int v = 0; v < 8; ++v) {
            int m = 8 * half + v;
            redS[wave * 16 + m] = bestS[v];
            redJ[wave * 16 + m] = bestJ[v];
        }
    }
    __syncthreads();

    // Threads 0..15 combine the 4 wave candidates per row, finish the loss.
    if (tid < 16) {
        const int m = tid;
        float bs = redS[m];
        int   bj = redJ[m];
#pragma unroll
        for (int w = 1; w < 4; ++w) {
            float s2 = redS[w * 16 + m];
            int   j2 = redJ[w * 16 + m];
            if (s2 < bs || (s2 == bs && j2 < bj)) { bs = s2; bj = j2; }
        }
        float d;
        if (bj == 0x7fffffff) {
            // Entire row masked out: argmin returns 0 -> dist to column 0.
            const float* xi = sXc + (rowBase + m) * 4;
            const float* x0 = sXc;
            float dot = xi[0] * x0[0] + xi[1] * x0[1] + xi[2] * x0[2];
            d = sqrtf(fmaxf(sSq[rowBase + m] + sSq[0] - 2.0f * dot, 0.0f));
        } else {
            d = sqrtf(bs);               // score-1 == dist at argmin
        }
        float val = tanhf(d) * inv_C;    // CONTACT_THRESH = 1
        // Reduce the 16 row contributions (lanes 0..15, masks stay in-range).
#pragma unroll
        for (int off = 1; off < 16; off <<= 1)
            val += __shfl_xor(val, off, 32);
        if (tid == 0) atomicAdd(out, val);
    }
}

extern "C" void kernel_launch(void* const* d_in, const int* in_sizes, int n_in,
                              void* d_out, int out_size, void* d_ws, size_t ws_size,
                              hipStream_t stream) {
    const int*           pc      = (const int*)d_in[0];            // [C] contact ids
    const float*         verts   = (const float*)d_in[1];          // [1,N,3] f32
    const unsigned char* geomask = (const unsigned char*)d_in[2];  // [N,N] bool bytes
    float*               out     = (float*)d_out;                  // scalar f32

    const int C = in_sizes[0];          // 1024
    const int N = in_sizes[1] / 3;      // 6890

    float4* Xc  = (float4*)d_ws;
    float*  sqn = (float*)((char*)d_ws + sizeof(float4) * (size_t)C);

    gather_contacts_kernel<<<(C + 255) / 256, 256, 0, stream>>>(pc, verts, Xc, sqn, out, C);

    // One 4-wave block per 16-row tile; each wave keeps EXEC all-1s at WMMA.
    contact_argmin_kernel<<<C / 16, 128, 0, stream>>>(
        pc, geomask, Xc, sqn, out, C, N, 1.0f / (float)C);
}